// Decoder_87608742904219
// MI455X (gfx1250) — compile-verified
//
#include <hip/hip_runtime.h>
#include <math.h>

// Problem sizes (match reference)
#define H 1024
#define V 50257
#define S 4096

typedef __attribute__((ext_vector_type(2))) float v2f;
typedef __attribute__((ext_vector_type(8))) float v8f;

// Workspace layout (float offsets). Total = 41984 floats ~= 168 KB.
#define OFF_H      0                   // 1024 : h_new copy
#define OFF_T      1024                // 1024 : t = attn_W^T @ h_new
#define OFF_BH     2048                // 1    : attn_b . h_new
#define OFF_STATS  2080                // 2    : [logit max, log(sum exp)]
#define OFF_SCORES 4096                // 4096 : raw attention scores
#define OFF_CTXP   8192                // 32*1024 : context partial sums
#define OFF_CTX    (8192 + 32*1024)    // 1024 : context vector

__device__ __forceinline__ float sigmoidf_(float x) { return 1.0f / (1.0f + expf(-x)); }

// ---------------------------------------------------------------------------
// K1: LSTM cell. gates = emb[word] @ W_ih^T + b_ih + h0 @ W_hh^T + b_hh
// One block per hidden index j; block computes the 4 gate rows {j, H+j, 2H+j, 3H+j}.
// Streams 33.5 MB of W_ih/W_hh.
// ---------------------------------------------------------------------------
__global__ void k_lstm(const int* __restrict__ word, const float* __restrict__ h0,
                       const float* __restrict__ c0, const float* __restrict__ emb,
                       const float* __restrict__ W_ih, const float* __restrict__ W_hh,
                       const float* __restrict__ b_ih, const float* __restrict__ b_hh,
                       float* __restrict__ out, float* __restrict__ ws)
{
    __shared__ float sx[H];
    __shared__ float sh[H];
    __shared__ float red[256];
    __shared__ float gsum[4];
    const int j = blockIdx.x;     // 0..1023
    const int t = threadIdx.x;    // 0..255
    const int w = word[0];
    for (int i = t; i < H; i += 256) { sx[i] = emb[(size_t)w * H + i]; sh[i] = h0[i]; }
    __syncthreads();
    for (int g = 0; g < 4; ++g) {
        const int row = g * H + j;
        const float* wi = W_ih + (size_t)row * H;
        const float* wh = W_hh + (size_t)row * H;
        float acc = 0.0f;
        #pragma unroll 4
        for (int k = t; k < H; k += 256) acc += wi[k] * sx[k] + wh[k] * sh[k];
        red[t] = acc; __syncthreads();
        for (int sft = 128; sft > 0; sft >>= 1) {
            if (t < sft) red[t] += red[t + sft];
            __syncthreads();
        }
        if (t == 0) gsum[g] = red[0] + b_ih[row] + b_hh[row];
        __syncthreads();
    }
    if (t == 0) {
        float ig = sigmoidf_(gsum[0]);
        float fg = sigmoidf_(gsum[1]);
        float gg = tanhf(gsum[2]);
        float og = sigmoidf_(gsum[3]);
        float cn = fg * c0[j] + ig * gg;
        float hn = og * tanhf(cn);
        out[V + j]     = hn;   // h_new output
        out[V + H + j] = cn;   // c_new output
        ws[OFF_H + j]  = hn;
    }
}

// ---------------------------------------------------------------------------
// K2: t[k] = sum_r attn_W[r,k] * h[r]   (blocks 0..3), bh = attn_b . h (block 4)
// Row-contiguous (coalesced) sweep over attn_W (4 MB).
// ---------------------------------------------------------------------------
__global__ void k_attn_t(const float* __restrict__ attn_W, const float* __restrict__ attn_b,
                         float* __restrict__ ws)
{
    __shared__ float sh[H];
    __shared__ float red[256];
    const int t = threadIdx.x;
    for (int i = t; i < H; i += 256) sh[i] = ws[OFF_H + i];
    __syncthreads();
    if (blockIdx.x == 4) {
        float acc = 0.0f;
        for (int r = t; r < H; r += 256) acc += attn_b[r] * sh[r];
        red[t] = acc; __syncthreads();
        for (int sft = 128; sft > 0; sft >>= 1) {
            if (t < sft) red[t] += red[t + sft];
            __syncthreads();
        }
        if (t == 0) ws[OFF_BH] = red[0];
        return;
    }
    const int k = blockIdx.x * 256 + t;
    float acc = 0.0f;
    #pragma unroll 8
    for (int r = 0; r < H; ++r) acc += attn_W[(size_t)r * H + k] * sh[r];
    ws[OFF_T + k] = acc;
}

// ---------------------------------------------------------------------------
// K3: scores[s] = enc[s,:] . t + bh.  One wave32 per encoder row (16 MB stream).
// ---------------------------------------------------------------------------
__global__ void k_scores(const float* __restrict__ enc, float* __restrict__ ws)
{
    __shared__ float st[H];
    const int t = threadIdx.x;
    for (int i = t; i < H; i += 256) st[i] = ws[OFF_T + i];
    __syncthreads();
    const int wave = t >> 5, lane = t & 31;
    const int s = blockIdx.x * 8 + wave;   // 0..4095
    const float* row = enc + (size_t)s * H;
    float acc = 0.0f;
    #pragma unroll 8
    for (int k = lane; k < H; k += 32) acc += row[k] * st[k];
    for (int off = 16; off > 0; off >>= 1) acc += __shfl_down(acc, off, 32);
    if (lane == 0) ws[OFF_SCORES + s] = acc + ws[OFF_BH];
}

// ---------------------------------------------------------------------------
// K4: softmax over S=4096 scores -> attn_w written to d_out[V+2H ..)
// ---------------------------------------------------------------------------
__global__ void k_softmax(float* __restrict__ out, float* __restrict__ ws)
{
    __shared__ float red[1024];
    __shared__ float Ms, Ss;
    const int t = threadIdx.x;
    float loc[4];
    float m = -INFINITY;
    #pragma unroll
    for (int i = 0; i < 4; ++i) { loc[i] = ws[OFF_SCORES + i * 1024 + t]; m = fmaxf(m, loc[i]); }
    red[t] = m; __syncthreads();
    for (int sft = 512; sft > 0; sft >>= 1) {
        if (t < sft) red[t] = fmaxf(red[t], red[t + sft]);
        __syncthreads();
    }
    if (t == 0) Ms = red[0];
    __syncthreads();
    float ssum = 0.0f;
    #pragma unroll
    for (int i = 0; i < 4; ++i) { loc[i] = expf(loc[i] - Ms); ssum += loc[i]; }
    red[t] = ssum; __syncthreads();
    for (int sft = 512; sft > 0; sft >>= 1) {
        if (t < sft) red[t] += red[t + sft];
        __syncthreads();
    }
    if (t == 0) Ss = red[0];
    __syncthreads();
    const float inv = 1.0f / Ss;
    #pragma unroll
    for (int i = 0; i < 4; ++i) out[V + 2 * H + i * 1024 + t] = loc[i] * inv;
}

// ---------------------------------------------------------------------------
// K5: context partials: ctxp[p,k] = sum_{s in chunk p} attn_w[s]*enc[s,k]
// grid (4 k-tiles, 32 s-chunks); enc is L2-resident after K3.
// ---------------------------------------------------------------------------
__global__ void k_ctx_part(const float* __restrict__ enc, const float* __restrict__ out,
                           float* __restrict__ ws)
{
    const int k  = blockIdx.x * 256 + threadIdx.x;
    const int s0 = blockIdx.y * 128;
    const float* aw = out + V + 2 * H;
    float acc = 0.0f;
    #pragma unroll 4
    for (int s = s0; s < s0 + 128; ++s) acc += aw[s] * enc[(size_t)s * H + k];
    ws[OFF_CTXP + blockIdx.y * H + k] = acc;
}

__global__ void k_ctx_reduce(float* __restrict__ ws)
{
    const int k = blockIdx.x * 256 + threadIdx.x;
    float acc = 0.0f;
    #pragma unroll
    for (int p = 0; p < 32; ++p) acc += ws[OFF_CTXP + p * H + k];
    ws[OFF_CTX + k] = acc;
}

// ---------------------------------------------------------------------------
// K6: logits[v] = feat . out_W[v,:] + out_b[v]  via V_WMMA_F32_16X16X4_F32.
// feat = [h_new | context] (2048), staged in LDS.
// Each wave owns 16 rows. A (16x4, documented layout): lane L streams b64 loads
// of out_W[v0+L%16, k0+2*(L/16) .. +1]. B (4x16) broadcasts feat[k0..k0+3] into
// all 16 N-columns via a ds_load_b64 (vgpr0 = K0/K2 halves, vgpr1 = K1/K3).
// D[m,n] is then n-invariant; lanes 0 and 16 hold the 16 logits in acc[0..7].
// This is the 412 MB stream that sets the kernel's runtime.
// ---------------------------------------------------------------------------
__global__ void __launch_bounds__(256) k_logits(const float* __restrict__ out_W,
                                                const float* __restrict__ out_b,
                                                float* __restrict__ out,
                                                const float* __restrict__ ws)
{
    __shared__ float feat[2 * H];
    const int t = threadIdx.x;
    for (int i = t; i < H; i += 256) { feat[i] = ws[OFF_H + i]; feat[H + i] = ws[OFF_CTX + i]; }
    __syncthreads();

    const int wave  = t >> 5;
    const int lane  = t & 31;
    const int v0    = blockIdx.x * 128 + wave * 16;   // first of this wave's 16 rows
    const int m     = lane & 15;                      // row within tile (A: M = lane%16)
    const int khalf = lane >> 4;                      // 0: K0/K1, 1: K2/K3

    int vr = v0 + m;                                  // clamp so OOB rows read safely
    if (vr > V - 1) vr = V - 1;
    const float* arow = out_W + (size_t)vr * (2 * H) + 2 * khalf;
    const float* bptr = feat + 2 * khalf;

    v8f acc = {};
    for (int k0 = 0; k0 < 2 * H; k0 += 16) {
        #pragma unroll
        for (int u = 0; u < 4; ++u) {
            v2f a = *(const v2f*)(arow + k0 + 4 * u);   // global_load_b64, A-layout native
            v2f b = *(const v2f*)(bptr + k0 + 4 * u);   // ds_load_b64 broadcast
            acc = __builtin_amdgcn_wmma_f32_16x16x4_f32(
                      false, a, false, b, (short)0, acc, false, false);
        }
    }
    // D layout: VGPR g -> lanes 0-15: M=g (N=lane); lanes 16-31: M=g+8.
    if (lane == 0) {
        #pragma unroll
        for (int g = 0; g < 8; ++g) { int v = v0 + g;     if (v < V) out[v] = acc[g] + out_b[v]; }
    } else if (lane == 16) {
        #pragma unroll
        for (int g = 0; g < 8; ++g) { int v = v0 + 8 + g; if (v < V) out[v] = acc[g] + out_b[v]; }
    }
}

// ---------------------------------------------------------------------------
// K7: log-sum-exp stats over the V logits (in d_out). 200 KB, single block.
// ---------------------------------------------------------------------------
__global__ void k_lse(const float* __restrict__ out, float* __restrict__ ws)
{
    __shared__ float red[1024];
    __shared__ float Ms;
    const int t = threadIdx.x;
    float m = -INFINITY;
    for (int v = t; v < V; v += 1024) m = fmaxf(m, out[v]);
    red[t] = m; __syncthreads();
    for (int sft = 512; sft > 0; sft >>= 1) {
        if (t < sft) red[t] = fmaxf(red[t], red[t + sft]);
        __syncthreads();
    }
    if (t == 0) Ms = red[0];
    __syncthreads();
    float ssum = 0.0f;
    for (int v = t; v < V; v += 1024) ssum += expf(out[v] - Ms);
    red[t] = ssum; __syncthreads();
    for (int sft = 512; sft > 0; sft >>= 1) {
        if (t < sft) red[t] += red[t + sft];
        __syncthreads();
    }
    if (t == 0) { ws[OFF_STATS] = Ms; ws[OFF_STATS + 1] = logf(red[0]); }
}

// K8: out[v] = logits[v] - max - log(sum exp)
__global__ void k_finalize(float* __restrict__ out, const float* __restrict__ ws)
{
    const int v = blockIdx.x * 256 + threadIdx.x;
    if (v < V) out[v] -= (ws[OFF_STATS] + ws[OFF_STATS + 1]);
}

// ---------------------------------------------------------------------------
extern "C" void kernel_launch(void* const* d_in, const int* in_sizes, int n_in,
                              void* d_out, int out_size, void* d_ws, size_t ws_size,
                              hipStream_t stream)
{
    (void)in_sizes; (void)n_in; (void)out_size; (void)ws_size;
    const int*   word   = (const int*)  d_in[0];
    const float* h0     = (const float*)d_in[1];
    const float* c0     = (const float*)d_in[2];
    const float* enc    = (const float*)d_in[3];
    const float* emb    = (const float*)d_in[4];
    const float* W_ih   = (const float*)d_in[5];
    const float* W_hh   = (const float*)d_in[6];
    const float* b_ih   = (const float*)d_in[7];
    const float* b_hh   = (const float*)d_in[8];
    const float* attn_W = (const float*)d_in[9];
    const float* attn_b = (const float*)d_in[10];
    const float* out_W  = (const float*)d_in[11];
    const float* out_b  = (const float*)d_in[12];
    float* out = (float*)d_out;
    float* ws  = (float*)d_ws;

    k_lstm      <<<H,            256, 0, stream>>>(word, h0, c0, emb, W_ih, W_hh, b_ih, b_hh, out, ws);
    k_attn_t    <<<5,            256, 0, stream>>>(attn_W, attn_b, ws);
    k_scores    <<<S / 8,        256, 0, stream>>>(enc, ws);
    k_softmax   <<<1,           1024, 0, stream>>>(out, ws);
    k_ctx_part  <<<dim3(4, 32),  256, 0, stream>>>(enc, out, ws);
    k_ctx_reduce<<<4,            256, 0, stream>>>(ws);
    k_logits    <<<(V + 127)/128,256, 0, stream>>>(out_W, out_b, out, ws);
    k_lse       <<<1,           1024, 0, stream>>>(out, ws);
    k_finalize  <<<(V + 255)/256,256, 0, stream>>>(out, ws);
}